// GCN_48610439856259
// MI455X (gfx1250) — compile-verified
//
#include <hip/hip_runtime.h>

// GCN forward for MI455X (gfx1250, wave32). fp32 WMMA (v_wmma_f32_16x16x4_f32)
// for the three dense GEMMs; K-pair-interleaved LDS-staged weights so every
// B fragment is a single ds_load_b64; atomic-f32 edge scatter.

typedef __attribute__((ext_vector_type(2))) float v2f;
typedef __attribute__((ext_vector_type(8))) float v8f;

#define GCN_NFEAT   128
#define GCN_HIDDEN  64
#define GCN_NCLASS  16

// ---------------------------------------------------------------- degree ----
__global__ void gcn_deg_init(float* deg, int n) {
  int i = blockIdx.x * blockDim.x + threadIdx.x;
  if (i < n) deg[i] = 1.0f;  // self-loop contributes 1
}

__global__ void gcn_deg_accum(const long long* __restrict__ dst, float* deg, int nE) {
  int e = blockIdx.x * blockDim.x + threadIdx.x;
  if (e < nE) atomicAdd(&deg[(int)dst[e]], 1.0f);
}

__global__ void gcn_rsqrt(float* deg, int n) {
  int i = blockIdx.x * blockDim.x + threadIdx.x;
  if (i < n) deg[i] = rsqrtf(deg[i]);  // deg >= 1 always (self loop)
}

// ------------------------------------------------------------- WMMA GEMM ----
// C[M x NDIM] = op(A)[M x KDIM] @ W[KDIM x NDIM] (+ bias), op = relu if RELU_A.
// One wave computes a 16-row stripe (16 x NDIM) with f32 16x16x4 WMMA.
// A-frag (16x4 f32): lanes 0-15 hold K=k,k+1; lanes 16-31 hold K=k+2,k+3 (row = lane&15).
// B-frag (4x16 f32): mirrored.  C/D: VGPR r -> M = r + (lane>=16 ? 8 : 0), N = lane&15.
// W is staged in LDS K-pair interleaved: Wp[((k>>1)*NDIM + n)*2 + (k&1)] = W[k][n],
// so each B fragment is one aligned 8-byte ds_load_b64 into an even VGPR pair.
template <int KDIM, int NDIM, bool RELU_A, bool ADD_BIAS>
__global__ __launch_bounds__(128) void gcn_gemm_wmma(
    const float* __restrict__ A, const float* __restrict__ W,
    const float* __restrict__ bias, float* __restrict__ C, int M) {
  __shared__ float Wlds[KDIM * NDIM];
  for (int i = threadIdx.x; i < KDIM * NDIM; i += blockDim.x) {
    int k = i / NDIM, n = i - k * NDIM;
    Wlds[(((k >> 1) * NDIM) + n) * 2 + (k & 1)] = W[i];
  }
  __syncthreads();

  const int lane = threadIdx.x & 31;
  const int wave = threadIdx.x >> 5;
  const int tiles_m = M >> 4;  // M divisible by 16 for this problem
  const int tile_m = blockIdx.x * (blockDim.x >> 5) + wave;
  if (tile_m >= tiles_m) return;  // wave-uniform: EXEC stays all-ones for WMMA

  const int half = lane >> 4;   // 0 or 1
  const int r    = lane & 15;   // row within tile / column within tile

  constexpr int NT = NDIM / 16;
  v8f zero = {};
  v8f acc[NT];
#pragma unroll
  for (int t = 0; t < NT; ++t) acc[t] = zero;

  const float* __restrict__ Ar = A + ((long)tile_m * 16 + r) * KDIM;

#pragma unroll 4
  for (int k = 0; k < KDIM; k += 4) {
    const int kk = k + half * 2;                    // even -> 8B aligned
    v2f a = *(const v2f*)(Ar + kk);                 // global_load_b64
    if (RELU_A) { a.x = fmaxf(a.x, 0.0f); a.y = fmaxf(a.y, 0.0f); }
    const int p = kk >> 1;                          // K-pair index
#pragma unroll
    for (int t = 0; t < NT; ++t) {
      v2f b = *(const v2f*)(&Wlds[(p * NDIM + t * 16 + r) * 2]);  // ds_load_b64
      acc[t] = __builtin_amdgcn_wmma_f32_16x16x4_f32(
          /*neg_a=*/false, a, /*neg_b=*/false, b,
          /*c_mod=*/(short)0, acc[t], /*reuse_a=*/false, /*reuse_b=*/false);
    }
  }

  const int mb = half * 8;
#pragma unroll
  for (int t = 0; t < NT; ++t) {
    if constexpr (ADD_BIAS) {
      const float bv = bias[t * 16 + r];
#pragma unroll
      for (int rr = 0; rr < 8; ++rr)
        C[((long)tile_m * 16 + mb + rr) * NDIM + t * 16 + r] = acc[t][rr] + bv;
    } else {
#pragma unroll
      for (int rr = 0; rr < 8; ++rr)
        C[((long)tile_m * 16 + mb + rr) * NDIM + t * 16 + r] = acc[t][rr];
    }
  }
}

// --------------------------------------------------- aggregation (scatter) --
// agg[i,f] = h[i,f]*dinv[i]^2 + bias[f]   (self-loop message + bias)
__global__ void gcn_agg_init(const float* __restrict__ h, const float* __restrict__ dinv,
                             const float* __restrict__ bias, float* __restrict__ agg, int n) {
  long idx = (long)blockIdx.x * blockDim.x + threadIdx.x;
  long total = (long)n * GCN_HIDDEN;
  if (idx < total) {
    int i = (int)(idx >> 6);   // / 64
    int f = (int)(idx & 63);
    float di = dinv[i];
    agg[idx] = h[idx] * di * di + bias[f];
  }
}

// agg[dst, :] += h[src, :] * dinv[src]*dinv[dst]  — 16 threads/edge, 4 floats each
__global__ void gcn_scatter(const long long* __restrict__ src, const long long* __restrict__ dst,
                            const float* __restrict__ dinv, const float* __restrict__ h,
                            float* agg, int nE) {
  long idx = (long)blockIdx.x * blockDim.x + threadIdx.x;
  long total = (long)nE * 16;
  if (idx >= total) return;
  int e  = (int)(idx >> 4);
  int f4 = (int)(idx & 15) * 4;
  int s = (int)src[e];
  int d = (int)dst[e];
  float w = dinv[s] * dinv[d];
  const float* __restrict__ hp = h + (long)s * GCN_HIDDEN + f4;
  float* ap = agg + (long)d * GCN_HIDDEN + f4;
  atomicAdd(ap + 0, hp[0] * w);
  atomicAdd(ap + 1, hp[1] * w);
  atomicAdd(ap + 2, hp[2] * w);
  atomicAdd(ap + 3, hp[3] * w);
}

// ----------------------------------------------------------------- softmax --
__global__ void gcn_softmax16(float* out, int n) {
  int i = blockIdx.x * blockDim.x + threadIdx.x;
  if (i >= n) return;
  float* row = out + (long)i * GCN_NCLASS;
  float v[GCN_NCLASS];
  float m = -3.4e38f;
#pragma unroll
  for (int c = 0; c < GCN_NCLASS; ++c) { v[c] = row[c]; m = fmaxf(m, v[c]); }
  float ssum = 0.0f;
#pragma unroll
  for (int c = 0; c < GCN_NCLASS; ++c) { v[c] = __expf(v[c] - m); ssum += v[c]; }
  float inv = 1.0f / ssum;
#pragma unroll
  for (int c = 0; c < GCN_NCLASS; ++c) row[c] = v[c] * inv;
}

// ------------------------------------------------------------------ launch --
static inline int ceil_div(long a, int b) { return (int)((a + b - 1) / b); }

extern "C" void kernel_launch(void* const* d_in, const int* in_sizes, int n_in,
                              void* d_out, int out_size, void* d_ws, size_t ws_size,
                              hipStream_t stream) {
  (void)n_in; (void)out_size; (void)ws_size;
  const float*      x    = (const float*)d_in[0];
  const long long*  ei   = (const long long*)d_in[1];  // int64 [2, E]
  const float*      W1   = (const float*)d_in[2];
  const float*      b1   = (const float*)d_in[3];
  const float*      W2   = (const float*)d_in[4];
  const float*      b2   = (const float*)d_in[5];
  const float*      Wout = (const float*)d_in[6];
  const float*      bout = (const float*)d_in[7];

  const int N = in_sizes[0] / GCN_NFEAT;  // 50000 (divisible by 16)
  const int E = in_sizes[1] / 2;          // 800000
  const long long* src = ei;
  const long long* dst = ei + E;

  // workspace: dinv[N] | bufA[N*64] | bufB[N*64]   (~26 MB)
  float* dinv = (float*)d_ws;
  float* bufA = dinv + N;
  float* bufB = bufA + (long)N * GCN_HIDDEN;

  // degree -> dinv
  gcn_deg_init <<<ceil_div(N, 256), 256, 0, stream>>>(dinv, N);
  gcn_deg_accum<<<ceil_div(E, 256), 256, 0, stream>>>(dst, dinv, E);
  gcn_rsqrt    <<<ceil_div(N, 256), 256, 0, stream>>>(dinv, N);

  const int tiles   = N / 16;               // 3125
  const int gblocks = ceil_div(tiles, 4);   // 4 waves / block
  const int aggthr  = ceil_div((long)N * GCN_HIDDEN, 256);
  const int scatthr = ceil_div((long)E * 16, 256);

  // layer 1: h1 = x @ W1 ; agg1 = scatter(norm * h1[src]) + b1
  gcn_gemm_wmma<GCN_NFEAT, GCN_HIDDEN, false, false>
      <<<gblocks, 128, 0, stream>>>(x, W1, nullptr, bufA, N);
  gcn_agg_init<<<aggthr, 256, 0, stream>>>(bufA, dinv, b1, bufB, N);
  gcn_scatter <<<scatthr, 256, 0, stream>>>(src, dst, dinv, bufA, bufB, E);

  // layer 2: h2 = relu(agg1) @ W2 ; agg2 = scatter + b2
  gcn_gemm_wmma<GCN_HIDDEN, GCN_HIDDEN, true, false>
      <<<gblocks, 128, 0, stream>>>(bufB, W2, nullptr, bufA, N);
  gcn_agg_init<<<aggthr, 256, 0, stream>>>(bufA, dinv, b2, bufB, N);
  gcn_scatter <<<scatthr, 256, 0, stream>>>(src, dst, dinv, bufA, bufB, E);

  // output: logits = relu(agg2) @ Wout + bout ; softmax rows of 16
  gcn_gemm_wmma<GCN_HIDDEN, GCN_NCLASS, true, true>
      <<<gblocks, 128, 0, stream>>>(bufB, Wout, bout, (float*)d_out, N);
  gcn_softmax16<<<ceil_div(N, 256), 256, 0, stream>>>((float*)d_out, N);
}